// PhysicsAttention_38139309589244
// MI455X (gfx1250) — compile-verified
//
#include <hip/hip_runtime.h>
#include <math.h>

typedef float v2f __attribute__((ext_vector_type(2)));
typedef float v8f __attribute__((ext_vector_type(8)));

constexpr int E   = 32;    // embed dim
constexpr int Bb  = 64;    // batch
constexpr int Ll  = 2048;  // seq len
constexpr int PSF = 21;    // PSF template length
constexpr int KT  = 128;   // keys per LDS tile
constexpr int KSTR = 36;   // LDS row stride (floats) for K tile (float4-aligned, conflict-free)
constexpr int VSTR = KT + 4;  // LDS row stride for transposed V tile
constexpr int PSTR = 20;   // LDS row stride for P staging

__device__ __forceinline__ int imin_(int a, int b) { return a < b ? a : b; }

// ---------------------------------------------------------------------------
// Build 21x21 Toeplitz Gaussian table and bilinear-resize index/frac tables.
// ---------------------------------------------------------------------------
__global__ __launch_bounds__(256) void psf_setup_kernel(float* __restrict__ topl,
                                                        int* __restrict__ midx,
                                                        float* __restrict__ mfrac) {
  __shared__ float t[PSF];
  __shared__ float ssum;
  const int tid = threadIdx.x;
  if (tid < PSF) {
    float xx = (float)(tid - 10);
    t[tid] = __expf(-0.5f * (xx * 0.5f) * (xx * 0.5f));
  }
  __syncthreads();
  if (tid == 0) {
    float s = 0.f;
    for (int i = 0; i < PSF; i++) s += t[i];
    ssum = s;
  }
  __syncthreads();
  if (tid < PSF) t[tid] /= ssum;
  __syncthreads();
  for (int i = tid; i < PSF * PSF; i += 256) {
    int r = i / PSF, c = i % PSF;
    int d = r - c; if (d < 0) d = -d;
    topl[i] = t[d];
  }
  const float scale = (float)PSF / (float)Ll;
  for (int i = tid; i < Ll; i += 256) {
    float src = ((float)i + 0.5f) * scale - 0.5f;
    src = fmaxf(src, 0.0f);                 // torch clamps negative source coords
    int i0 = (int)floorf(src);
    float fr = src - (float)i0;
    if (i0 > PSF - 1) i0 = PSF - 1;
    midx[i] = i0;
    mfrac[i] = fr;
  }
}

// ---------------------------------------------------------------------------
// QKV projection: x[B,E,L] -> q,k,v [B,L,E]; q pre-scaled by 1/sqrt(E).
// ---------------------------------------------------------------------------
__global__ __launch_bounds__(256) void qkv_kernel(const float* __restrict__ x,
                                                  const float* __restrict__ W,
                                                  const float* __restrict__ bia,
                                                  float* __restrict__ q,
                                                  float* __restrict__ k,
                                                  float* __restrict__ v) {
  __shared__ float sW[3 * E * E];
  __shared__ float sB[3 * E];
  const int tid = threadIdx.x;
  for (int i = tid; i < 3 * E * E; i += 256) sW[i] = W[i];
  for (int i = tid; i < 3 * E; i += 256) sB[i] = bia[i];
  __syncthreads();
  const int b = blockIdx.y;
  const int l = blockIdx.x * 256 + tid;
  float xv[E];
#pragma unroll
  for (int c = 0; c < E; c++) xv[c] = x[((size_t)b * E + c) * Ll + l];  // coalesced on l
  const float qs = 0.17677669529663687f;  // 1/sqrt(32)
  const size_t ro = ((size_t)b * Ll + l) * E;
#pragma unroll 4
  for (int e = 0; e < E; e++) {
    float aq = sB[e], ak = sB[E + e], av = sB[2 * E + e];
#pragma unroll
    for (int c = 0; c < E; c++) {
      aq += sW[e * E + c] * xv[c];
      ak += sW[(E + e) * E + c] * xv[c];
      av += sW[(2 * E + e) * E + c] * xv[c];
    }
    q[ro + e] = aq * qs;
    k[ro + e] = ak;
    v[ro + e] = av;
  }
}

// ---------------------------------------------------------------------------
// Fused flash attention with multiplicative PSF mask.
// 256 threads = 8 wave32; each wave owns 16 query rows; WG shares K/V tiles.
// fp32 WMMA 16x16x4 for QK^T, P*V, and the softmax row-sum (ones-column trick).
// ---------------------------------------------------------------------------
__global__ __launch_bounds__(256) void attn_kernel(const float* __restrict__ q,
                                                   const float* __restrict__ k,
                                                   const float* __restrict__ v,
                                                   const float* __restrict__ topl,
                                                   const int* __restrict__ midx,
                                                   const float* __restrict__ mfrac,
                                                   float* __restrict__ outO) {
  __shared__ float sT[PSF * PSF];
  __shared__ float sK[KT * KSTR];       // K tile, row-major [key][feat]
  __shared__ float sVt[E * VSTR];       // V tile, transposed [feat][key]
  __shared__ float sP[8][16 * PSTR];    // per-wave P staging (C-layout -> A-layout)
  __shared__ float sR[8][16 * PSF];     // per-wave row-lerped mask table [m][jj]
  __shared__ int   sJi[KT];             // column mask indices for this key block
  __shared__ float sJf[KT];             // column mask fracs

  const int tid = threadIdx.x;
  const int wave = tid >> 5;
  const int lane = tid & 31;
  const int half = lane >> 4;  // lane half selects K-offset in A/B layouts
  const int ln = lane & 15;
  const int b = blockIdx.y;
  const int qbase = blockIdx.x * 128 + wave * 16;

  for (int i = tid; i < PSF * PSF; i += 256) sT[i] = topl[i];
  __syncthreads();

  // Wave-private row-lerped mask table: sR[wave][m][jj] = lerp_i(T)[row m][jj]
  {
    float* rt = &sR[wave][0];
    for (int i = lane; i < 16 * PSF; i += 32) {
      int m = i / PSF, jj = i - m * PSF;
      int row = qbase + m;
      int i0 = midx[row];
      float fi = mfrac[row];
      int i1 = imin_(i0 + 1, PSF - 1);
      rt[i] = (1.0f - fi) * sT[i0 * PSF + jj] + fi * sT[i1 * PSF + jj];
    }
  }

  // Q rows in A-layout registers: chunk s holds K=4s..4s+3.
  v2f aq[8];
  {
    const float* qr = q + ((size_t)b * Ll + qbase + ln) * E;
#pragma unroll
    for (int s = 0; s < 8; s++) {
      int f = 4 * s + 2 * half;
      aq[s].x = qr[f];
      aq[s].y = qr[f + 1];
    }
  }

  float mrun[8];
#pragma unroll
  for (int r = 0; r < 8; r++) mrun[r] = -1e30f;
  v8f o0 = {0.f, 0.f, 0.f, 0.f, 0.f, 0.f, 0.f, 0.f};  // feats 0..15
  v8f o1 = {0.f, 0.f, 0.f, 0.f, 0.f, 0.f, 0.f, 0.f};  // feats 16..31
  v8f o2 = {0.f, 0.f, 0.f, 0.f, 0.f, 0.f, 0.f, 0.f};  // virtual ones-column: row sums

  // Constant B operand for the ones-column accumulator: V[n][32+ln] = (ln==0)
  v2f bones;
  bones.x = (ln == 0) ? 1.0f : 0.0f;
  bones.y = bones.x;

  const float* rtab = &sR[wave][8 * half * PSF];  // rows for this lane half

  for (int kb = 0; kb < Ll; kb += KT) {
    __syncthreads();
    // Cooperative staging: K row-major, V transposed, column mask params.
    {
      const float4* kg = (const float4*)(k + ((size_t)b * Ll + kb) * E);
      const float4* vg = (const float4*)(v + ((size_t)b * Ll + kb) * E);
#pragma unroll
      for (int it = 0; it < (KT * E / 4) / 256; ++it) {
        int idx = it * 256 + tid;
        int row = idx >> 3, f4 = idx & 7;
        float4 kk = kg[idx];
        float4 vv = vg[idx];
        *(float4*)&sK[row * KSTR + f4 * 4] = kk;
        sVt[(4 * f4 + 0) * VSTR + row] = vv.x;
        sVt[(4 * f4 + 1) * VSTR + row] = vv.y;
        sVt[(4 * f4 + 2) * VSTR + row] = vv.z;
        sVt[(4 * f4 + 3) * VSTR + row] = vv.w;
      }
      for (int i = tid; i < KT; i += 256) {
        sJi[i] = midx[kb + i];
        sJf[i] = mfrac[kb + i];
      }
    }
    __syncthreads();

#pragma unroll
    for (int t = 0; t < KT / 16; t++) {
      // ---- S = Q * K^T (16x16), K-dim = 32 in 8 steps of 4 ----
      v8f s = {0.f, 0.f, 0.f, 0.f, 0.f, 0.f, 0.f, 0.f};
      const int krow = (t * 16 + ln) * KSTR;
#pragma unroll
      for (int st = 0; st < 8; st++) {
        v2f bk;  // B[kdim][n] = Ktile[n][kdim]; lane: n=ln, kdim=4st+2*half
        int f = 4 * st + 2 * half;
        bk.x = sK[krow + f];
        bk.y = sK[krow + f + 1];
        s = __builtin_amdgcn_wmma_f32_16x16x4_f32(false, aq[st], false, bk,
                                                  (short)0, s, false, false);
      }

      // ---- multiplicative PSF mask: col-lerp of precomputed row-lerped table ----
      const int jl = t * 16 + ln;
      const int j0 = sJi[jl];
      const float fj = sJf[jl];
      const int j1 = imin_(j0 + 1, PSF - 1);
      const float w0 = 1.0f - fj;
      float sr[8];
#pragma unroll
      for (int r = 0; r < 8; r++) {
        float mv = w0 * rtab[r * PSF + j0] + fj * rtab[r * PSF + j1];
        sr[r] = s[r] * mv;
      }

      // ---- online softmax: per-row max reduce + rescale accumulators ----
      float pv[8], al[8];
#pragma unroll
      for (int r = 0; r < 8; r++) {
        float x = sr[r];
        x = fmaxf(x, __shfl_xor(x, 1, 32));
        x = fmaxf(x, __shfl_xor(x, 2, 32));
        x = fmaxf(x, __shfl_xor(x, 4, 32));
        x = fmaxf(x, __shfl_xor(x, 8, 32));
        float mn = fmaxf(mrun[r], x);
        al[r] = __expf(mrun[r] - mn);
        mrun[r] = mn;
        pv[r] = __expf(sr[r] - mn);
      }
#pragma unroll
      for (int r = 0; r < 8; r++) {
        o0[r] *= al[r];
        o1[r] *= al[r];
        o2[r] *= al[r];
      }

      // ---- stage P through wave-private LDS (C-layout -> A-layout) ----
      float* mp = &sP[wave][0];
#pragma unroll
      for (int r = 0; r < 8; r++) mp[(r + 8 * half) * PSTR + ln] = pv[r];

      // ---- O += P * [V | 1] (three 16x16 accumulators, K-dim = 16 in 4 steps)
      //      o2's B operand is the constant ones-column pattern: row sums for free.
#pragma unroll
      for (int st = 0; st < 4; st++) {
        v2f ap, b0, b1;
        int c0 = 4 * st + 2 * half;
        ap.x = mp[ln * PSTR + c0];
        ap.y = mp[ln * PSTR + c0 + 1];
        int nloc = t * 16 + c0;  // key index within KT tile
        b0.x = sVt[ln * VSTR + nloc];
        b0.y = sVt[ln * VSTR + nloc + 1];
        b1.x = sVt[(16 + ln) * VSTR + nloc];
        b1.y = sVt[(16 + ln) * VSTR + nloc + 1];
        o0 = __builtin_amdgcn_wmma_f32_16x16x4_f32(false, ap, false, b0,
                                                   (short)0, o0, false, false);
        o1 = __builtin_amdgcn_wmma_f32_16x16x4_f32(false, ap, false, b1,
                                                   (short)0, o1, false, false);
        o2 = __builtin_amdgcn_wmma_f32_16x16x4_f32(false, ap, false, bones,
                                                   (short)0, o2, false, false);
      }
    }
  }

  // Normalize and write O[b, qbase+m, e]. Row sum lives in o2 column 0, which
  // every lane of a half needs: broadcast within the 16-lane half via shfl.
  float* orow = outO + ((size_t)b * Ll + qbase) * E;
#pragma unroll
  for (int r = 0; r < 8; r++) {
    float lsum = o2[r];
    // broadcast column-0 value (lane ln==0 of each half) to all lanes of the half
    lsum = __shfl(lsum, half * 16, 32);
    float inv = 1.0f / lsum;
    int m = r + 8 * half;
    orow[(size_t)m * E + ln] = o0[r] * inv;
    orow[(size_t)m * E + 16 + ln] = o1[r] * inv;
  }
}

// ---------------------------------------------------------------------------
// Output projection: O[B,L,E] -> out[B,E,L] (coalesced writes on l).
// ---------------------------------------------------------------------------
__global__ __launch_bounds__(256) void outproj_kernel(const float* __restrict__ O,
                                                      const float* __restrict__ W,
                                                      const float* __restrict__ bia,
                                                      float* __restrict__ out) {
  __shared__ float sW[E * E];
  __shared__ float sB[E];
  const int tid = threadIdx.x;
  for (int i = tid; i < E * E; i += 256) sW[i] = W[i];
  for (int i = tid; i < E; i += 256) sB[i] = bia[i];
  __syncthreads();
  const int b = blockIdx.y;
  const int l = blockIdx.x * 256 + tid;
  float ov[E];
  const float4* op = (const float4*)(O + ((size_t)b * Ll + l) * E);
#pragma unroll
  for (int f4 = 0; f4 < E / 4; f4++) {
    float4 t4 = op[f4];
    ov[f4 * 4 + 0] = t4.x;
    ov[f4 * 4 + 1] = t4.y;
    ov[f4 * 4 + 2] = t4.z;
    ov[f4 * 4 + 3] = t4.w;
  }
#pragma unroll 4
  for (int e = 0; e < E; e++) {
    float acc = sB[e];
#pragma unroll
    for (int f = 0; f < E; f++) acc += sW[e * E + f] * ov[f];
    out[((size_t)b * E + e) * Ll + l] = acc;
  }
}

// ---------------------------------------------------------------------------
extern "C" void kernel_launch(void* const* d_in, const int* in_sizes, int n_in,
                              void* d_out, int out_size, void* d_ws, size_t ws_size,
                              hipStream_t stream) {
  (void)in_sizes; (void)n_in; (void)out_size; (void)ws_size;
  const float* x    = (const float*)d_in[0];
  const float* Wqkv = (const float*)d_in[1];
  const float* bqkv = (const float*)d_in[2];
  const float* Wout = (const float*)d_in[3];
  const float* bout = (const float*)d_in[4];
  float* out = (float*)d_out;

  char* ws = (char*)d_ws;
  const size_t tsz = (size_t)Bb * Ll * E * sizeof(float);  // 16.78 MB per tensor
  float* q    = (float*)(ws + 0 * tsz);
  float* kk   = (float*)(ws + 1 * tsz);
  float* vv   = (float*)(ws + 2 * tsz);
  float* O    = (float*)(ws + 3 * tsz);
  float* topl = (float*)(ws + 4 * tsz);
  int*   midx = (int*)(ws + 4 * tsz + 512 * sizeof(float));
  float* mfrac= (float*)(ws + 4 * tsz + 512 * sizeof(float) + (size_t)Ll * sizeof(int));

  psf_setup_kernel<<<1, 256, 0, stream>>>(topl, midx, mfrac);
  qkv_kernel<<<dim3(Ll / 256, Bb), 256, 0, stream>>>(x, Wqkv, bqkv, q, kk, vv);
  attn_kernel<<<dim3(Ll / 128, Bb), 256, 0, stream>>>(q, kk, vv, topl, midx, mfrac, O);
  outproj_kernel<<<dim3(Ll / 256, Bb), 256, 0, stream>>>(O, Wout, bout, out);
}